// Test_25331717111922
// MI455X (gfx1250) — compile-verified
//
#include <hip/hip_runtime.h>

// Bilinear table interpolation, 2^24 points into a 2048x512 f32 table.
// Bandwidth-bound: ~192 MB streaming (r, z, out) -> ~8.2 us floor at 23.3 TB/s.
// Table (4 MB) is kept L2-resident by using non-temporal policy on all
// streaming traffic; table gathers are L2 hits.

constexpr int NR_G = 2048;
constexpr int NZ_G = 512;   // row stride of the table (z is contiguous)

typedef float v4f __attribute__((ext_vector_type(4)));

// 8-byte pair at 4-byte alignment: t[i,j] and t[i,j+1] are adjacent.
struct __attribute__((packed, aligned(4))) fpair { float lo, hi; };

__device__ __forceinline__ float bilerp(const float* __restrict__ t,
                                        float x, float y) {
    // H = 1.0, RGRID0 = ZGRID0 = 0.0  =>  grid coords == physical coords
    int i = (int)floorf(x);
    int j = (int)floorf(y);
    i = min(max(i, 0), NR_G - 2);
    j = min(max(j, 0), NZ_G - 2);
    float wx = x - (float)i;
    float wy = y - (float)j;

    const float* p = t + (i << 9) + j;          // i*NZ + j, NZ = 512
    fpair t0 = *(const fpair*)(p);              // t00, t01  (row i)
    fpair t1 = *(const fpair*)(p + NZ_G);       // t10, t11  (row i+1)

    // lerp along z in each row, then along r
    float a = fmaf(wy, t0.hi - t0.lo, t0.lo);
    float b = fmaf(wy, t1.hi - t1.lo, t1.lo);
    return fmaf(wx, b - a, a);
}

extern "C" __global__ void __launch_bounds__(256)
interp_bilinear_kernel(const float* __restrict__ r,
                       const float* __restrict__ z,
                       const float* __restrict__ table,
                       float* __restrict__ out,
                       int n4) {
    const v4f* __restrict__ r4 = (const v4f*)r;
    const v4f* __restrict__ z4 = (const v4f*)z;
    v4f* __restrict__ o4 = (v4f*)out;

    int tid    = blockIdx.x * blockDim.x + threadIdx.x;
    int stride = gridDim.x * blockDim.x;

    for (int k = tid; k < n4; k += stride) {
        // gfx1250 global_prefetch_b8: pull next stride's streaming chunk
        // toward L2 while we work on this one (speculative; OOB is dropped).
        __builtin_prefetch((const void*)(r4 + k + stride), 0, 1);
        __builtin_prefetch((const void*)(z4 + k + stride), 0, 1);

        // Non-temporal B128 loads: read-once data must not evict the 4 MB
        // table from L2.
        v4f rv = __builtin_nontemporal_load(r4 + k);
        v4f zv = __builtin_nontemporal_load(z4 + k);

        v4f o;
        o.x = bilerp(table, rv.x, zv.x);
        o.y = bilerp(table, rv.y, zv.y);
        o.z = bilerp(table, rv.z, zv.z);
        o.w = bilerp(table, rv.w, zv.w);

        // Non-temporal B128 store: write-once result.
        __builtin_nontemporal_store(o, o4 + k);
    }
}

extern "C" void kernel_launch(void* const* d_in, const int* in_sizes, int n_in,
                              void* d_out, int out_size, void* d_ws, size_t ws_size,
                              hipStream_t stream) {
    // setup_inputs order: r, z, timetable, timetable_grad_r, timetable_grad_z
    const float* r     = (const float*)d_in[0];
    const float* z     = (const float*)d_in[1];
    const float* table = (const float*)d_in[2];
    float* out = (float*)d_out;

    int n  = in_sizes[0];      // 16,777,216
    int n4 = n >> 2;           // float4 elements

    const int threads = 256;   // 8 wave32 per block
    const int blocks  = 4096;  // grid-stride: 4 iterations/thread at 2^24 pts

    interp_bilinear_kernel<<<blocks, threads, 0, stream>>>(r, z, table, out, n4);
}